// AutoInt_26766236188985
// MI455X (gfx1250) — compile-verified
//
#include <hip/hip_runtime.h>

typedef __attribute__((ext_vector_type(2))) float v2f;
typedef __attribute__((ext_vector_type(8))) float v8f;

#define BATCH   16384
#define NSP     26
#define INDIM   39
#define FEAT    39
#define FPAD    48
#define EMB     8
#define HEADS   2
#define HDIM    4
#define NLAY    3
#define VOC     11
#define EMBOUT  (FEAT*EMB)   /* 312 */
#define HS      9            /* padded row stride for 48x8 buffers (odd -> conflict-free) */
#define SS      49           /* padded row stride for 48x48 score buffer */
#define KSTEPS  10           /* att@V K-steps: only k < 40 can be nonzero (39 cols + 1 zero) */
#define LDS_FLOATS (5*FPAD*HS + FPAD*SS)

static __device__ __forceinline__ v8f wmma4(v2f a, v2f b, v8f c) {
    // D = A(16x4,f32) * B(4x16,f32) + C(16x16,f32)
    return __builtin_amdgcn_wmma_f32_16x16x4_f32(false, a, false, b, (short)0, c,
                                                 false, false);
}

__global__ __launch_bounds__(32, 1) void autoint_fused(
    const float* __restrict__ x,
    const float* __restrict__ emb_tables,
    const float* __restrict__ dense_W,
    const float* __restrict__ dense_b,
    const float* __restrict__ Wq,
    const float* __restrict__ Wk,
    const float* __restrict__ Wv,
    const float* __restrict__ Wr,
    const float* __restrict__ fcW,
    const float* __restrict__ fcb,
    float* __restrict__ out_emb,
    float* __restrict__ out_prob)
{
    __shared__ float lds[LDS_FLOATS];
    const int lane  = threadIdx.x & 31;
    const int nn    = lane & 15;      // N (or A-row) position
    const int lh    = lane >> 4;      // wave half
    const int khalf = lh * 2;         // K-pair owned by this half

    float* hb = lds;                          // h       48 x 8 (stride 9)
    float* qb = hb + FPAD*HS;                 // q
    float* kb = qb + FPAD*HS;                 // k
    float* vb = kb + FPAD*HS;                 // v
    float* rb = vb + FPAD*HS;                 // h @ Wres (residual term)
    float* sb = rb + FPAD*HS;                 // softmaxed scores 48 x 48 (stride 49)

    const int b = blockIdx.x;                 // one sample per 32-thread block (1 wave)

    // lane-constant branchless selectors (pointer/index cndmask, no EXEC branching)
    const float* wqk_base = (nn < 8) ? Wq : Wk;
    const float* wvr_base = (nn < 8) ? Wv : Wr;
    const int    wcol     = nn & 7;
    float* dQK = (nn < 8) ? qb : kb;
    float* dVR = (nn < 8) ? vb : rb;
    const float vmask = (nn < HDIM) ? 1.0f : 0.0f;
    const int   vcolo = nn & (HDIM - 1);      // clamped column for att@V B frag

    // ---- preload all packed projection B-fragments for all layers into registers
    v2f bqk[NLAY][2], bvr[NLAY][2];
    #pragma unroll
    for (int l = 0; l < NLAY; ++l) {
        #pragma unroll
        for (int s = 0; s < 2; ++s) {
            #pragma unroll
            for (int v = 0; v < 2; ++v) {
                int kk = s*4 + khalf + v;
                bqk[l][s][v] = wqk_base[(l*EMB + kk)*EMB + wcol];
                bvr[l][s][v] = wvr_base[(l*EMB + kk)*EMB + wcol];
            }
        }
    }

    // ---- phase 0: h = [embeddings ; dense outer-product feats], zero-pad rows 39..47
    for (int t = lane; t < FPAD*EMB; t += 32) {
        int f = t >> 3, e = t & 7;
        float val = 0.0f;
        if (f < NSP) {
            int idx = (int)x[b*INDIM + f];    // sparse values >= 0 -> no min-shift
            idx = idx < 0 ? 0 : (idx > VOC-1 ? VOC-1 : idx);
            val = emb_tables[(f*VOC + idx)*EMB + e];
        } else if (f < FEAT) {
            int dd = f - NSP;
            val = fmaf(x[b*INDIM + f], dense_W[dd*EMB + e], dense_b[dd*EMB + e]);
        }
        hb[f*HS + e] = val;
    }
    __syncthreads();   // single-wave workgroup: barrier is a NOP, kept as ordering fence

    #pragma unroll
    for (int l = 0; l < NLAY; ++l) {
        // ---- projections: q|k and v|r (K=8 -> 2 k-steps each)
        #pragma unroll
        for (int mt = 0; mt < 3; ++mt) {
            v8f dqk = {}; v8f dvr = {};
            #pragma unroll
            for (int s = 0; s < 2; ++s) {
                v2f a;
                a[0] = hb[(mt*16 + nn)*HS + s*4 + khalf + 0];
                a[1] = hb[(mt*16 + nn)*HS + s*4 + khalf + 1];
                dqk = wmma4(a, bqk[l][s], dqk);
                dvr = wmma4(a, bvr[l][s], dvr);
            }
            #pragma unroll
            for (int vv = 0; vv < 8; ++vv) {
                int m2 = mt*16 + vv + 8*lh;
                dQK[m2*HS + wcol] = dqk[vv];
                dVR[m2*HS + wcol] = dvr[vv];
            }
        }
        __syncthreads();

        #pragma unroll
        for (int hd = 0; hd < HEADS; ++hd) {
            const int c0 = hd * HDIM;
            // ---- scores Q·K^T (K=4 == head_dim -> 1 WMMA per 16x16 tile) + fused softmax
            #pragma unroll
            for (int mi = 0; mi < 3; ++mi) {
                v2f a;
                a[0] = qb[(mi*16 + nn)*HS + c0 + khalf + 0];
                a[1] = qb[(mi*16 + nn)*HS + c0 + khalf + 1];
                v8f s0 = {}, s1 = {}, s2 = {};
                v2f bf;
                bf[0] = kb[(0*16 + nn)*HS + c0 + khalf + 0];
                bf[1] = kb[(0*16 + nn)*HS + c0 + khalf + 1];
                s0 = wmma4(a, bf, s0);
                bf[0] = kb[(1*16 + nn)*HS + c0 + khalf + 0];
                bf[1] = kb[(1*16 + nn)*HS + c0 + khalf + 1];
                s1 = wmma4(a, bf, s1);
                bf[0] = kb[(2*16 + nn)*HS + c0 + khalf + 0];
                bf[1] = kb[(2*16 + nn)*HS + c0 + khalf + 1];
                s2 = wmma4(a, bf, s2);
                // D layout: row i = mi*16 + vv + 8*lh, col j = 16*nj + nn.
                // softmax over j: 16-lane shfl reduction (stays inside each half).
                #pragma unroll
                for (int vv = 0; vv < 8; ++vv) {
                    float x0 = s0[vv], x1 = s1[vv];
                    float x2 = (nn < FEAT - 32) ? s2[vv] : -3.0e38f;  // mask j>=39
                    float mx = fmaxf(x0, fmaxf(x1, x2));
                    #pragma unroll
                    for (int m = 1; m < 16; m <<= 1) mx = fmaxf(mx, __shfl_xor(mx, m, 32));
                    float e0 = __expf(x0 - mx);
                    float e1 = __expf(x1 - mx);
                    float e2 = __expf(x2 - mx);        // underflows to exactly 0 when masked
                    float sum = e0 + e1 + e2;
                    #pragma unroll
                    for (int m = 1; m < 16; m <<= 1) sum += __shfl_xor(sum, m, 32);
                    float inv = 1.0f / sum;
                    int i = mi*16 + vv + 8*lh;
                    sb[i*SS +  0 + nn] = e0 * inv;
                    sb[i*SS + 16 + nn] = e1 * inv;
                    sb[i*SS + 32 + nn] = e2 * inv;     // exactly 0 for padded j
                }
            }
            __syncthreads();

            // ---- att @ V (K-steps cover k < 40 only; sb cols >= 40 are all zero),
            //      fused +residual and ReLU into h
            const int vcol = c0 + vcolo;
            #pragma unroll
            for (int mt = 0; mt < 3; ++mt) {
                v8f acc = {};
                #pragma unroll
                for (int s = 0; s < KSTEPS; ++s) {
                    v2f a, bf;
                    a[0] = sb[(mt*16 + nn)*SS + s*4 + khalf + 0];
                    a[1] = sb[(mt*16 + nn)*SS + s*4 + khalf + 1];
                    int kk = s*4 + khalf;
                    bf[0] = vb[(kk + 0)*HS + vcol] * vmask;   // branchless masked load
                    bf[1] = vb[(kk + 1)*HS + vcol] * vmask;
                    acc = wmma4(a, bf, acc);
                }
                if (nn < HDIM) {
                    #pragma unroll
                    for (int vv = 0; vv < 8; ++vv) {
                        int m2 = mt*16 + vv + 8*lh;
                        float val = acc[vv] + rb[m2*HS + c0 + nn];
                        hb[m2*HS + c0 + nn] = fmaxf(val, 0.0f);
                    }
                }
            }
            __syncthreads();
        }
    }

    // ---- outputs: emb (B x 312) then 2-class softmax (B x 2)
    for (int t = lane; t < EMBOUT; t += 32)
        out_emb[(size_t)b*EMBOUT + t] = hb[(t >> 3)*HS + (t & 7)];

    float p0 = 0.0f, p1 = 0.0f;
    for (int t = lane; t < EMBOUT; t += 32) {
        float e = hb[(t >> 3)*HS + (t & 7)];
        p0 = fmaf(e, fcW[t*2 + 0], p0);
        p1 = fmaf(e, fcW[t*2 + 1], p1);
    }
    #pragma unroll
    for (int m = 1; m < 32; m <<= 1) {
        p0 += __shfl_xor(p0, m, 32);
        p1 += __shfl_xor(p1, m, 32);
    }
    if (lane == 0) {
        p0 += fcb[0]; p1 += fcb[1];
        float mx = fmaxf(p0, p1);
        float e0 = __expf(p0 - mx), e1 = __expf(p1 - mx);
        float inv = 1.0f / (e0 + e1);
        out_prob[(size_t)b*2 + 0] = e0 * inv;
        out_prob[(size_t)b*2 + 1] = e1 * inv;
    }
}

extern "C" void kernel_launch(void* const* d_in, const int* in_sizes, int n_in,
                              void* d_out, int out_size, void* d_ws, size_t ws_size,
                              hipStream_t stream) {
    (void)in_sizes; (void)n_in; (void)out_size; (void)d_ws; (void)ws_size;
    const float* x    = (const float*)d_in[0];
    const float* emb  = (const float*)d_in[1];
    const float* dW   = (const float*)d_in[2];
    const float* db   = (const float*)d_in[3];
    const float* Wq   = (const float*)d_in[4];
    const float* Wk   = (const float*)d_in[5];
    const float* Wv   = (const float*)d_in[6];
    const float* Wr   = (const float*)d_in[7];
    const float* fcW  = (const float*)d_in[8];
    const float* fcb  = (const float*)d_in[9];
    float* out_emb  = (float*)d_out;
    float* out_prob = out_emb + (size_t)BATCH * EMBOUT;

    dim3 grid(BATCH), block(32);   // one sample per single-wave workgroup
    hipLaunchKernelGGL(autoint_fused, grid, block, 0, stream,
                       x, emb, dW, db, Wq, Wk, Wv, Wr, fcW, fcb, out_emb, out_prob);
}